// StructureAwareLoss_50096498540700
// MI455X (gfx1250) — compile-verified
//
#include <hip/hip_runtime.h>
#include <stdint.h>

// Problem constants (must match reference)
#define B_      512
#define L_      16384
#define NCLS    3
#define WCONT   0.05f

// Tiling
#define TILE     2048            // timesteps per workgroup
#define NTHREADS 256
#define STEPS    (TILE / NTHREADS)   // 8 sequential steps per thread
#define KCH      (L_ / TILE)         // 8 chunks per sequence
#define NBLK     (B_ * KCH)          // 4096 blocks
#define NWAVES   (NTHREADS / 32)     // 8 waves (wave32)
#define NEGINF   (-1.0e30f)

// Workspace layout (float offsets). All slots written (not accumulated)
// every call -> deterministic, no init pass needed.
#define WS_MAT   0                    // NBLK * 9 chunk log-matrices
#define WS_SCORE (WS_MAT  + NBLK*9)   // NBLK  partial numerator scores
#define WS_MASKC (WS_SCORE + NBLK)    // NBLK  partial mask counts
#define WS_FOCAL (WS_MASKC + NBLK)    // NBLK  partial focal sums
#define WS_NVAL  (WS_FOCAL + NBLK)    // NBLK  partial valid counts
#define WS_TRC   (WS_NVAL  + NBLK)    // NBLK  partial transition sums
#define WS_NTR   (WS_TRC   + NBLK)    // NBLK  partial transition counts
#define WS_LLH   (WS_NTR   + NBLK)    // B_    per-sequence log-likelihood

typedef __attribute__((ext_vector_type(2))) float v2f;
typedef __attribute__((ext_vector_type(8))) float v8f;

__device__ __forceinline__ float lse3(float a, float b, float c) {
  float m = fmaxf(a, fmaxf(b, c));
  return m + __logf(__expf(a - m) + __expf(b - m) + __expf(c - m));
}

// CDNA5 async global->LDS copy (ASYNCcnt-tracked, no VGPR staging)
__device__ __forceinline__ void async_b128(uint32_t lds_off, const void* gptr) {
  asm volatile("global_load_async_to_lds_b128 %0, %1, off"
               :: "v"(lds_off), "v"(gptr) : "memory");
}
__device__ __forceinline__ void wait_async0() {
  asm volatile("s_wait_asynccnt 0" ::: "memory");
}

// Exact f32 wave-sum via V_WMMA_F32_16X16X4_F32 with a ones B-matrix:
// A vgpr0 = per-lane value, vgpr1 = 0  =>  D[M][N] = v(M) + v(M+16) for all N.
// Lanes n and n^16 together hold one full column of D, so summing a lane's
// 8 D regs and adding the xor-16 partner yields the exact 32-lane total.
__device__ __forceinline__ float wmma_wave_sum(float v) {
  v2f a; a[0] = v;   a[1] = 0.f;
  v2f bo; bo[0] = 1.f; bo[1] = 1.f;
  v8f c = {};
  v8f d = __builtin_amdgcn_wmma_f32_16x16x4_f32(
      /*neg_a=*/false, a, /*neg_b=*/false, bo,
      /*c_mod=*/(short)0, c, /*reuse_a=*/false, /*reuse_b=*/false);
  float t = ((d[0] + d[1]) + (d[2] + d[3])) + ((d[4] + d[5]) + (d[6] + d[7]));
  t += __shfl_xor(t, 16, 32);
  return t;   // uniform across the wave
}

__global__ __launch_bounds__(NTHREADS)
void saloss_tile_kernel(const float* __restrict__ logits,
                        const int* __restrict__ targets,
                        const unsigned char* __restrict__ mask,
                        const float* __restrict__ transg,
                        float* __restrict__ ws)
{
  __shared__ __align__(16) float         smL[TILE * 3];
  __shared__ __align__(16) int           smT[TILE];
  __shared__ __align__(16) unsigned char smM[TILE];
  __shared__ float smMat[NTHREADS * 9];
  __shared__ float redW[6 * NWAVES];
  __shared__ int shBtag, shBpred;

  const int tid = threadIdx.x;
  const int blk = blockIdx.x;
  const int b   = blk / KCH;
  const int kc  = blk % KCH;
  const int tile0 = kc * TILE;

  const float*         gL = logits  + ((size_t)b * L_ + tile0) * 3;
  const int*           gT = targets + (size_t)b * L_ + tile0;
  const unsigned char* gM = mask    + (size_t)b * L_ + tile0;

  const uint32_t ldsL = (uint32_t)(uintptr_t)smL;
  const uint32_t ldsT = (uint32_t)(uintptr_t)smT;
  const uint32_t ldsM = (uint32_t)(uintptr_t)smM;

  // ---- async stage tile into LDS (24KB logits + 8KB targets + 2KB mask) ----
  #pragma unroll
  for (int i = 0; i < (TILE * 3 * 4) / (NTHREADS * 16); ++i) {   // 6 iters
    uint32_t off = (uint32_t)(i * NTHREADS + tid) * 16u;
    async_b128(ldsL + off, (const char*)gL + off);
  }
  #pragma unroll
  for (int i = 0; i < (TILE * 4) / (NTHREADS * 16); ++i) {       // 2 iters
    uint32_t off = (uint32_t)(i * NTHREADS + tid) * 16u;
    async_b128(ldsT + off, (const char*)gT + off);
  }
  if (tid < TILE / 16) {                                          // 128 lanes
    uint32_t off = (uint32_t)tid * 16u;
    async_b128(ldsM + off, (const char*)gM + off);
  }

  // ---- tile-boundary state (tag & argmax-pred at t0-1) ----
  if (tid == 0) {
    int btag = 0, bpred = 0;
    if (tile0 > 0) {
      const float* pL = logits + ((size_t)b * L_ + tile0 - 1) * 3;
      float e0 = pL[0], e1 = pL[1], e2 = pL[2];
      float bb = e0; bpred = 0;
      if (e1 > bb) { bb = e1; bpred = 1; }
      if (e2 > bb) { bpred = 2; }
      int tr = targets[(size_t)b * L_ + tile0 - 1];
      btag = tr < 0 ? 0 : (tr > NCLS - 1 ? NCLS - 1 : tr);
    }
    shBtag = btag; shBpred = bpred;
  }

  // exp(trans) once per thread: turns the log-semiring step into 27 FMAs
  float trv[9], A[9];
  #pragma unroll
  for (int i = 0; i < 9; ++i) { trv[i] = transg[i]; A[i] = __expf(trv[i]); }

  wait_async0();
  __syncthreads();

  // ---- per-thread sequential subchunk [base, base+STEPS) ----
  const int base = tid * STEPS;
  int prevpred, prevtag;
  if (tid == 0) { prevpred = shBpred; prevtag = shBtag; }
  else {
    int tl = base - 1;
    float e0 = smL[tl*3], e1 = smL[tl*3+1], e2 = smL[tl*3+2];
    float bb = e0; prevpred = 0;
    if (e1 > bb) { bb = e1; prevpred = 1; }
    if (e2 > bb) { prevpred = 2; }
    int tr = smT[tl];
    prevtag = tr < 0 ? 0 : (tr > 2 ? 2 : tr);
  }

  // linear-domain running chunk product P (3x3), renorm log-sum in csum
  float p00=1.f,p01=0.f,p02=0.f, p10=0.f,p11=1.f,p12=0.f, p20=0.f,p21=0.f,p22=1.f;
  float csum = 0.f;
  float accF=0.f, accNV=0.f, accS=0.f, accMC=0.f, accTR=0.f, accNTR=0.f;

  #pragma unroll
  for (int s = 0; s < STEPS; ++s) {
    const int tl = base + s;
    const int tg = tile0 + tl;
    float e0 = smL[tl*3], e1 = smL[tl*3+1], e2 = smL[tl*3+2];
    int tgt = smT[tl];
    int mk  = smM[tl] ? 1 : 0;

    // shared softmax pieces (focal + CRF renorm share the 3 exps)
    float mx = fmaxf(e0, fmaxf(e1, e2));
    float x0 = __expf(e0 - mx), x1 = __expf(e1 - mx), x2 = __expf(e2 - mx);
    float se  = x0 + x1 + x2;
    float lse = __logf(se);

    int tgs   = tgt < 0 ? 0 : (tgt > 2 ? 2 : tgt);
    int valid = (mk && tgt >= 0 && tgt < NCLS) ? 1 : 0;
    float lt  = (tgs == 0 ? e0 : (tgs == 1 ? e1 : e2)) - mx;
    float xt  = (tgs == 0 ? x0 : (tgs == 1 ? x1 : x2));
    float ce  = lse - lt;
    float pt  = xt / se;
    float om  = 1.f - pt;
    if (valid) { accF += om * om * ce; accNV += 1.f; }
    accMC += (float)mk;

    int pred = 0; { float bb = e0;
      if (e1 > bb) { bb = e1; pred = 1; }
      if (e2 > bb) { pred = 2; } }

    if (tg >= 1) {
      if (valid) { accNTR += 1.f; if (pred != prevpred) accTR += 1.f; }
      if (mk) {
        accS += trv[prevtag * 3 + tgs] + (lt + mx);
        // P = P * (A .* w), w[j] = exp(em[j]-mx); identity if masked
        csum += mx;
        float n00 = x0*(p00*A[0] + p01*A[3] + p02*A[6]);
        float n01 = x1*(p00*A[1] + p01*A[4] + p02*A[7]);
        float n02 = x2*(p00*A[2] + p01*A[5] + p02*A[8]);
        float n10 = x0*(p10*A[0] + p11*A[3] + p12*A[6]);
        float n11 = x1*(p10*A[1] + p11*A[4] + p12*A[7]);
        float n12 = x2*(p10*A[2] + p11*A[5] + p12*A[8]);
        float n20 = x0*(p20*A[0] + p21*A[3] + p22*A[6]);
        float n21 = x1*(p20*A[1] + p21*A[4] + p22*A[7]);
        float n22 = x2*(p20*A[2] + p21*A[5] + p22*A[8]);
        p00=n00; p01=n01; p02=n02; p10=n10; p11=n11; p12=n12; p20=n20; p21=n21; p22=n22;
      }
    }
    prevpred = pred; prevtag = tgs;
  }

  // back to log domain for the associative combine
  {
    float pl[9] = {p00,p01,p02,p10,p11,p12,p20,p21,p22};
    #pragma unroll
    for (int i = 0; i < 9; ++i)
      smMat[tid*9 + i] = (pl[i] > 0.f) ? (__logf(pl[i]) + csum) : NEGINF;
  }

  // ---- WMMA-based exact f32 wave reduction of the 6 scalar partials ----
  // (EXEC is all-ones here: no divergence at this point.)
  {
    const int wave = tid >> 5;
    const int lane = tid & 31;
    float vals[6] = {accF, accNV, accS, accMC, accTR, accNTR};
    #pragma unroll
    for (int v = 0; v < 6; ++v) {
      float t = wmma_wave_sum(vals[v]);
      if (lane == 0) redW[v * NWAVES + wave] = t;
    }
  }
  __syncthreads();

  // ---- ordered binary-tree log-semiring matmul combine (256 -> 1) ----
  for (int off = 1; off < NTHREADS; off <<= 1) {
    const bool act = ((tid & (2*off - 1)) == 0);
    if (act) {
      float Lm[9], Rm[9];
      #pragma unroll
      for (int i = 0; i < 9; ++i) { Lm[i] = smMat[tid*9 + i]; Rm[i] = smMat[(tid+off)*9 + i]; }
      #pragma unroll
      for (int i = 0; i < 3; ++i)
        #pragma unroll
        for (int j = 0; j < 3; ++j)
          smMat[tid*9 + i*3 + j] =
            lse3(Lm[i*3+0] + Rm[0*3+j], Lm[i*3+1] + Rm[1*3+j], Lm[i*3+2] + Rm[2*3+j]);
    }
    __syncthreads();
  }

  if (tid == 0) {
    // ordered cross-wave combine of the WMMA wave sums (deterministic)
    float tot[6];
    #pragma unroll
    for (int v = 0; v < 6; ++v) {
      float s = 0.f;
      #pragma unroll
      for (int w = 0; w < NWAVES; ++w) s += redW[v * NWAVES + w];
      tot[v] = s;
    }
    ws[WS_FOCAL + blk] = tot[0];
    ws[WS_NVAL  + blk] = tot[1];
    ws[WS_SCORE + blk] = tot[2];
    ws[WS_MASKC + blk] = tot[3];
    ws[WS_TRC   + blk] = tot[4];
    ws[WS_NTR   + blk] = tot[5];
    #pragma unroll
    for (int i = 0; i < 9; ++i) ws[WS_MAT + (size_t)blk*9 + i] = smMat[i];
  }
}

// One thread per sequence: fold alpha0 through K chunk matrices, finish llh.
__global__ __launch_bounds__(256)
void saloss_finalizeA(const float* __restrict__ logits,
                      const int* __restrict__ targets,
                      const float* __restrict__ st,
                      const float* __restrict__ et,
                      float* __restrict__ ws)
{
  int b = blockIdx.x * blockDim.x + threadIdx.x;
  if (b >= B_) return;

  float a0 = st[0] + logits[(size_t)b*L_*3 + 0];
  float a1 = st[1] + logits[(size_t)b*L_*3 + 1];
  float a2 = st[2] + logits[(size_t)b*L_*3 + 2];

  float sc = 0.f, msum = 0.f;
  #pragma unroll
  for (int kk = 0; kk < KCH; ++kk) {
    int blk = b * KCH + kk;
    sc   += ws[WS_SCORE + blk];
    msum += ws[WS_MASKC + blk];
    const float* M = &ws[WS_MAT + (size_t)blk*9];
    float n0 = lse3(a0 + M[0], a1 + M[3], a2 + M[6]);
    float n1 = lse3(a0 + M[1], a1 + M[4], a2 + M[7]);
    float n2 = lse3(a0 + M[2], a1 + M[5], a2 + M[8]);
    a0 = n0; a1 = n1; a2 = n2;
  }

  int t0r = targets[(size_t)b*L_];
  int t0s = t0r < 0 ? 0 : (t0r > 2 ? 2 : t0r);
  sc += st[t0s] + logits[(size_t)b*L_*3 + t0s];

  int send = (int)(msum + 0.5f) - 1;
  send = send < 0 ? 0 : (send >= L_ ? L_ - 1 : send);
  int ltr = targets[(size_t)b*L_ + send];
  int lts = ltr < 0 ? 0 : (ltr > 2 ? 2 : ltr);

  float denom = lse3(a0 + et[0], a1 + et[1], a2 + et[2]);
  ws[WS_LLH + b] = (sc + et[lts]) - denom;
}

// Single block: deterministic tree-sum of all partials -> final scalar.
__global__ __launch_bounds__(1024)
void saloss_finalizeB(const float* __restrict__ ws, float* __restrict__ out)
{
  __shared__ float s5[5 * 1024];
  int tid = threadIdx.x;
  float f = 0.f, nv = 0.f, tr = 0.f, ntr = 0.f, l = 0.f;
  for (int i = tid; i < NBLK; i += 1024) {
    f   += ws[WS_FOCAL + i];
    nv  += ws[WS_NVAL  + i];
    tr  += ws[WS_TRC   + i];
    ntr += ws[WS_NTR   + i];
  }
  for (int i = tid; i < B_; i += 1024) l += ws[WS_LLH + i];
  s5[0*1024+tid] = f;  s5[1*1024+tid] = nv; s5[2*1024+tid] = tr;
  s5[3*1024+tid] = ntr; s5[4*1024+tid] = l;
  __syncthreads();
  for (int off = 512; off > 0; off >>= 1) {
    if (tid < off) {
      #pragma unroll
      for (int v = 0; v < 5; ++v) s5[v*1024 + tid] += s5[v*1024 + tid + off];
    }
    __syncthreads();
  }
  if (tid == 0) {
    float focal = s5[0*1024] / fmaxf(s5[1*1024], 1.f);
    float cont  = s5[2*1024] / fmaxf(s5[3*1024], 1.f);
    float crf   = -s5[4*1024] / (float)B_;
    out[0] = focal + crf + WCONT * cont;
  }
}

extern "C" void kernel_launch(void* const* d_in, const int* in_sizes, int n_in,
                              void* d_out, int out_size, void* d_ws, size_t ws_size,
                              hipStream_t stream)
{
  (void)in_sizes; (void)n_in; (void)out_size; (void)ws_size;
  const float*         logits  = (const float*)d_in[0];
  const int*           targets = (const int*)d_in[1];
  const unsigned char* mask    = (const unsigned char*)d_in[2];
  const float*         st      = (const float*)d_in[3];
  const float*         et      = (const float*)d_in[4];
  const float*         trs     = (const float*)d_in[5];
  float* ws  = (float*)d_ws;
  float* out = (float*)d_out;

  saloss_tile_kernel<<<NBLK, NTHREADS, 0, stream>>>(logits, targets, mask, trs, ws);
  saloss_finalizeA<<<(B_ + 255) / 256, 256, 0, stream>>>(logits, targets, st, et, ws);
  saloss_finalizeB<<<1, 1024, 0, stream>>>(ws, out);
}